// Colorcal_13907104104704
// MI455X (gfx1250) — compile-verified
//
#include <hip/hip_runtime.h>
#include <cstdint>

// out[b,c,h,w] = img[b,c,h,w] * weight[cam[b],c] + bias[cam[b],c]
// Streaming, memory-bound (402 MB @ 23.3 TB/s ~= 17 us floor).
// Data path: GLOBAL_LOAD_ASYNC_TO_LDS_B128 (4-stage pipeline, barrier-free,
// each lane reads back its own LDS slot) -> ds_load_b128 -> v_fma_f32 with
// SGPR scale/bias -> non-temporal global_store_b128.

typedef __attribute__((ext_vector_type(4))) float v4f;

constexpr int kH = 1024;
constexpr int kW = 1024;
constexpr int kC = 3;
constexpr int kPlaneF4 = (kH * kW) / 4;     // float4 per (b,c) plane = 262144
constexpr int kTPB = 256;                   // 8 wave32s
constexpr int kIter = 8;                    // float4 per thread
constexpr int kDepth = 4;                   // async pipeline stages
constexpr int kChunkF4 = kTPB * kIter;      // 2048 float4 per block
constexpr int kBlocksPerPlane = kPlaneF4 / kChunkF4;  // 128

__device__ __forceinline__ void async_copy_f4(const float* __restrict__ base,
                                              uint32_t byte_off,
                                              uint32_t lds_byte_addr) {
  // GVS mode: 64-bit SGPR base + 32-bit per-lane byte offset -> per-lane LDS addr.
  asm volatile("global_load_async_to_lds_b128 %0, %1, %2"
               :
               : "v"(lds_byte_addr), "v"(byte_off), "s"(base)
               : "memory");
}

__device__ __forceinline__ void wait_async_le(int n) {
  // n is compile-time constant after full unroll; folds to a single SOPP.
  if (n <= 0)      asm volatile("s_wait_asynccnt 0x0" ::: "memory");
  else if (n == 1) asm volatile("s_wait_asynccnt 0x1" ::: "memory");
  else if (n == 2) asm volatile("s_wait_asynccnt 0x2" ::: "memory");
  else             asm volatile("s_wait_asynccnt 0x3" ::: "memory");
}

__device__ __forceinline__ void wait_ds_zero() {
  // DS (DScnt) and async-LDS writes (ASYNCcnt) are mutually unordered in the
  // LDS pipeline; drain pending ds_loads before overwriting a recycled slot.
  asm volatile("s_wait_dscnt 0x0" ::: "memory");
}

__global__ __launch_bounds__(kTPB) void colorcal_kernel(
    const float* __restrict__ img,
    const int* __restrict__ cam_idx,
    const float* __restrict__ weight,
    const float* __restrict__ bias,
    float* __restrict__ out) {
  __shared__ v4f sbuf[kDepth][kTPB];

  const int plane = blockIdx.y;            // plane = b*3 + c, uniform per block
  const int b = plane / kC;
  const int c = plane - kC * b;
  const int cam = cam_idx[b];              // uniform -> scalar load
  const float scale = weight[cam * kC + c];
  const float shift = bias[cam * kC + c];

  const uint32_t tid = threadIdx.x;
  const uint32_t base_f4 =
      (uint32_t)plane * (uint32_t)kPlaneF4 +
      (uint32_t)blockIdx.x * (uint32_t)kChunkF4 + tid;

  // Low 32 bits of the flat shared-memory address == wave-relative LDS byte
  // offset (LDS aperture truncation, ISA 10.2).
  const uint32_t lds0 = (uint32_t)(uintptr_t)(void*)&sbuf[0][tid];

  // Prologue: fill the pipeline (kDepth-1 stages in flight, fresh slots).
#pragma unroll
  for (int p = 0; p < kDepth - 1; ++p) {
    async_copy_f4(img, (base_f4 + (uint32_t)(p * kTPB)) * 16u,
                  lds0 + (uint32_t)(p * kTPB * 16));
  }

  v4f* __restrict__ outf4 = (v4f*)out;

#pragma unroll
  for (int i = 0; i < kIter; ++i) {
    // Issue the stage kDepth-1 ahead; it recycles the slot whose ds_load
    // happened at iteration i-1, so drain DScnt first (steady-state free:
    // that data was already consumed by the adjacent fma).
    if (i + kDepth - 1 < kIter) {
      const int s = (i + kDepth - 1) % kDepth;
      if (i > 0) wait_ds_zero();
      async_copy_f4(img, (base_f4 + (uint32_t)((i + kDepth - 1) * kTPB)) * 16u,
                    lds0 + (uint32_t)(s * kTPB * 16));
    }
    // Async ops complete in order: stage i is done once the number of
    // outstanding ops drops to those issued after it.
    int rem = kIter - 1 - i;
    if (rem > kDepth - 1) rem = kDepth - 1;
    wait_async_le(rem);

    v4f v = sbuf[i % kDepth][tid];         // ds_load_b128
    v.x = __builtin_fmaf(v.x, scale, shift);
    v.y = __builtin_fmaf(v.y, scale, shift);
    v.z = __builtin_fmaf(v.z, scale, shift);
    v.w = __builtin_fmaf(v.w, scale, shift);

    // Written once, never re-read by this dispatch -> non-temporal store.
    __builtin_nontemporal_store(v, &outf4[base_f4 + (uint32_t)(i * kTPB)]);
  }
}

extern "C" void kernel_launch(void* const* d_in, const int* in_sizes, int n_in,
                              void* d_out, int out_size, void* d_ws,
                              size_t ws_size, hipStream_t stream) {
  const float* img    = (const float*)d_in[0];
  const int*   cam    = (const int*)d_in[1];
  const float* weight = (const float*)d_in[2];
  const float* bias   = (const float*)d_in[3];
  float*       out    = (float*)d_out;

  const int B = in_sizes[1];               // 16 samples
  dim3 grid(kBlocksPerPlane, B * kC, 1);   // (128, 48)
  colorcal_kernel<<<grid, kTPB, 0, stream>>>(img, cam, weight, bias, out);
}